// GraphLayerNorm_40578851012881
// MI455X (gfx1250) — compile-verified
//
#include <hip/hip_runtime.h>
#include <hip/hip_bf16.h>

// GraphLayerNorm for MI455X (gfx1250, wave32).
// N=262144 nodes, D=256 channels, B=1024 graphs, batch sorted.
// Memory-bound (AI ~0.45 flop/byte): 2-pass scheme, var = E[x^2]-mean^2.
//   Pass 1 (stats): V_WMMA_F32_16X16X4_F32 with all-ones B as the row-
//     reduction engine, 16 nodes/wave; loads stay RT so x populates L2.
//   Pass 2 (normalize): walks x in REVERSE order — x is 256MB, L2 is 192MB,
//     so the tail of x (MRU after pass 1) is still L2-resident; reading
//     tail-first converts most of the second x read into L2 hits.
//     x reads are last-use -> NT loads; out is write-once -> NT stores,
//     so neither evicts the resident part of x.

#define GN_N 262144
#define GN_D 256
#define GN_B 1024
#define GN_EPS 1e-05f

typedef __attribute__((ext_vector_type(2))) float v2f;
typedef __attribute__((ext_vector_type(4))) float v4f;   // native vector for NT builtins
typedef __attribute__((ext_vector_type(8))) float v8f;

#if __has_builtin(__builtin_amdgcn_wmma_f32_16x16x4_f32)
#define GN_HAVE_WMMA 1
#else
#define GN_HAVE_WMMA 0
#endif

__global__ void gn_zero_kernel(float* __restrict__ p, int n) {
    int i = blockIdx.x * blockDim.x + threadIdx.x;
    if (i < n) p[i] = 0.0f;
}

// One wave handles 16 nodes. A-fragment (16x4 f32, per ISA layout):
//   lane m in 0..15  -> row m, cols {k, k+1}
//   lane m in 16..31 -> row m-16, cols {k+2, k+3}
// B = all ones => D[m,n] = sum_k A[m,k] + C[m,n]; every column of D holds the
// row sum, so lane 0 / lane 16 read rows 0..7 / 8..15 out of the 8 acc VGPRs.
__global__ void gn_stats_kernel(const float* __restrict__ x,
                                const int* __restrict__ batch,
                                float* __restrict__ gsum,
                                float* __restrict__ gsumsq,
                                float* __restrict__ gcnt) {
    const int lane = threadIdx.x & 31;
    const int wave = threadIdx.x >> 5;
    const int waves_per_block = blockDim.x >> 5;
    const int node_base = (blockIdx.x * waves_per_block + wave) * 16;
    const int m = lane & 15;
    const int koff = (lane >> 4) << 1;  // 0 for lanes 0-15, 2 for lanes 16-31

    const float* row = x + (size_t)(node_base + m) * GN_D + koff;

#if GN_HAVE_WMMA
    v2f bones = {1.0f, 1.0f};
    v8f csum = {};
    v8f csq  = {};
    for (int k = 0; k < GN_D; k += 4) {
        v2f a  = *(const v2f*)(row + k);
        v2f a2 = a * a;
        csum = __builtin_amdgcn_wmma_f32_16x16x4_f32(
            false, a,  false, bones, (short)0, csum, false, false);
        csq  = __builtin_amdgcn_wmma_f32_16x16x4_f32(
            false, a2, false, bones, (short)0, csq,  false, false);
    }
    // Extract row sums: VGPR v, lanes 0-15 -> row v; lanes 16-31 -> row v+8.
    if (lane == 0 || lane == 16) {
        const int mbase = (lane >> 4) * 8;
        #pragma unroll
        for (int v = 0; v < 8; ++v) {
            const int node = node_base + mbase + v;
            const int g = batch[node];
            atomicAdd(&gsum[g],   csum[v]);
            atomicAdd(&gsumsq[g], csq[v]);
            atomicAdd(&gcnt[g],   1.0f);
        }
    }
#else
    // Fallback: per-lane partial sums; lane m and lane m+16 hold the two
    // halves of row m's reduction.
    float psum = 0.0f, psq = 0.0f;
    for (int k = 0; k < GN_D; k += 4) {
        v2f a = *(const v2f*)(row + k);
        psum += a.x + a.y;
        psq  += a.x * a.x + a.y * a.y;
    }
    float osum = __shfl(psum, lane + 16, 32);
    float osq  = __shfl(psq,  lane + 16, 32);
    if (lane < 16) {
        const int node = node_base + m;
        const int g = batch[node];
        atomicAdd(&gsum[g],   psum + osum);
        atomicAdd(&gsumsq[g], psq + osq);
        atomicAdd(&gcnt[g],   1.0f);
    }
#endif
}

__global__ void gn_finalize_kernel(const float* __restrict__ gsum,
                                   const float* __restrict__ gsumsq,
                                   const float* __restrict__ gcnt,
                                   float* __restrict__ gmean,
                                   float* __restrict__ ginv) {
    int g = blockIdx.x * blockDim.x + threadIdx.x;
    if (g < GN_B) {
        float c    = gcnt[g];
        float norm = fmaxf(c, 1.0f) * (float)GN_D;
        float mu   = gsum[g] / norm;
        float ex2  = gsumsq[g] / norm;
        float var  = fmaxf(ex2 - mu * mu, 0.0f);
        gmean[g] = mu;
        ginv[g]  = 1.0f / (sqrtf(var) + GN_EPS);  // eps added AFTER sqrt (ref)
    }
}

__global__ void gn_normalize_kernel(const float* __restrict__ x,
                                    const int* __restrict__ batch,
                                    const float* __restrict__ weight,
                                    const float* __restrict__ bias,
                                    const float* __restrict__ gmean,
                                    const float* __restrict__ ginv,
                                    float* __restrict__ out,
                                    int total4) {
    // Reverse traversal: block 0 handles the LAST chunk of x, which is the
    // most-recently-cached data from the stats pass (L2 tail reuse).
    const int base = total4 - (blockIdx.x + 1) * blockDim.x;  // grid divides exactly
    const int i    = base + (int)threadIdx.x;                 // float4 index

    const int node = i >> 6;        // D/4 = 64 float4 per node
    const int c4   = i & 63;
    const int g    = batch[node];
    const float mu = gmean[g];
    const float s  = ginv[g];
    // Last-use read of x: non-temporal so it doesn't evict still-resident x.
    const v4f xv = __builtin_nontemporal_load((const v4f*)x + i);
    const v4f wv = ((const v4f*)weight)[c4];
    const v4f bv = ((const v4f*)bias)[c4];
    v4f o = (xv - mu) * s * wv + bv;
    // Write-once output: non-temporal store, don't pollute L2.
    __builtin_nontemporal_store(o, (v4f*)out + i);
}

extern "C" void kernel_launch(void* const* d_in, const int* in_sizes, int n_in,
                              void* d_out, int out_size, void* d_ws, size_t ws_size,
                              hipStream_t stream) {
    const float* x      = (const float*)d_in[0];
    const int*   batch  = (const int*)d_in[1];
    const float* weight = (const float*)d_in[2];
    const float* bias   = (const float*)d_in[3];
    float* out = (float*)d_out;

    float* ws     = (float*)d_ws;
    float* gsum   = ws;              // [B]
    float* gsumsq = ws + GN_B;       // [B]
    float* gcnt   = ws + 2 * GN_B;   // [B]
    float* gmean  = ws + 3 * GN_B;   // [B]
    float* ginv   = ws + 4 * GN_B;   // [B]

    // Zero accumulators every launch (harness does not re-poison between
    // replays; this keeps kernel_launch deterministic & idempotent).
    gn_zero_kernel<<<(3 * GN_B + 255) / 256, 256, 0, stream>>>(ws, 3 * GN_B);

    // Stats: 4 waves/block, 16 nodes/wave -> 64 nodes/block. RT-cached loads
    // deliberately leave the tail of x resident in the 192MB L2.
    gn_stats_kernel<<<GN_N / 64, 128, 0, stream>>>(x, batch, gsum, gsumsq, gcnt);

    gn_finalize_kernel<<<GN_B / 256, 256, 0, stream>>>(gsum, gsumsq, gcnt,
                                                       gmean, ginv);

    // Normalize: one float4 per thread, reverse order for L2 tail reuse.
    const int total4 = GN_N * (GN_D / 4);
    gn_normalize_kernel<<<total4 / 256, 256, 0, stream>>>(x, batch, weight, bias,
                                                          gmean, ginv, out, total4);
}